// VectorQuantize_35304631173724
// MI455X (gfx1250) — compile-verified
//
#include <hip/hip_runtime.h>

typedef __attribute__((ext_vector_type(2))) float v2f;
typedef __attribute__((ext_vector_type(4))) float v4f;
typedef __attribute__((ext_vector_type(8))) float v8f;

#define EMB    256
#define NCODE  1024
#define LPOS   2048
#define NROWS  32768                 // 16 * 2048
#define ZOUT   (NROWS * EMB)         // 8388608 floats
#define DIFF_OFF ZOUT
#define IDX_OFF  (ZOUT + 1)

#define CHUNK   64                   // codes staged per LDS buffer
#define NCHUNK  (NCODE / CHUNK)      // 16
#define BROW    260                  // padded row stride (floats): conflict-free
#define BUFF    (CHUNK * BROW)       // floats per buffer

// ---------------------------------------------------------------------------
// Prep: bias[k] = -0.5 * ||W_k||^2 ; zero the diff accumulator (ws[0]).
// ---------------------------------------------------------------------------
__global__ __launch_bounds__(256) void vq_prep(const float* __restrict__ W,
                                               float* __restrict__ ws) {
  int k = blockIdx.x * blockDim.x + threadIdx.x;
  if (k == 0) ws[0] = 0.0f;
  if (k < NCODE) {
    const v4f* wr = (const v4f*)(W + (size_t)k * EMB);
    float s = 0.0f;
#pragma unroll 8
    for (int i = 0; i < EMB / 4; ++i) {
      v4f v = wr[i];
      s += v.x * v.x + v.y * v.y + v.z * v.z + v.w * v.w;
    }
    ws[1 + k] = -0.5f * s;
  }
}

// low 32 bits of a generic pointer to __shared__ == wave-relative LDS offset
__device__ __forceinline__ unsigned lds_off(const void* p) {
  return (unsigned)(unsigned long long)p;
}

// Issue async DMA of one 64-code chunk (64 rows x 256 f32) into an LDS buffer.
// 256 threads x 16 iterations x 16B = 64KB; LDS rows padded to 260 floats.
__device__ __forceinline__ void issue_chunk(const float* __restrict__ W,
                                            int g, unsigned ldsBase, int tid) {
  const float* src = W + (size_t)g * CHUNK * EMB;
#pragma unroll
  for (int i = 0; i < 16; ++i) {
    int q   = i * 256 + tid;        // 0..4095 float4 slots
    int row = q >> 6;               // 64 float4 per row
    int col = q & 63;
    unsigned     loff = ldsBase + (unsigned)row * (BROW * 4u) + (unsigned)col * 16u;
    const float* ga   = src + row * EMB + col * 4;
    asm volatile("global_load_async_to_lds_b128 %0, %1, off"
                 :: "v"(loff), "v"(ga) : "memory");
  }
}

// ---------------------------------------------------------------------------
// Main: per wave 16 rows x all 1024 codes via V_WMMA_F32_16X16X4_F32.
// Codebook double-buffered in LDS via async-to-LDS DMA; A tile in VGPRs.
// ---------------------------------------------------------------------------
__global__ __launch_bounds__(256) void vq_main(const float* __restrict__ X,
                                               const float* __restrict__ W,
                                               const float* __restrict__ bias,
                                               float* __restrict__ out,
                                               float* __restrict__ diff_acc) {
  __shared__ __align__(16) float ldsW[2 * BUFF];   // ~130 KB
  __shared__ int   kIdxS[8][16];
  __shared__ float redS[8];

  const int wave    = threadIdx.x >> 5;
  const int lane    = threadIdx.x & 31;
  const int M       = lane & 15;                 // A row / B column within tile
  const int off     = (lane < 16) ? 0 : 2;       // K sub-offset per ISA layout
  const int rowBase = blockIdx.x * 128 + wave * 16;
  const unsigned ldsW0 = lds_off(&ldsW[0]);

  // kick off DMA of chunk 0 while we load the A tile
  issue_chunk(W, 0, ldsW0, threadIdx.x);

  // --- A tile: 16 rows x 256 in WMMA 16x4 layout => contiguous float2/lane
  v2f a[64];
  {
    const float* arow = X + (size_t)(rowBase + M) * EMB + off;
#pragma unroll
    for (int c = 0; c < 64; ++c) a[c] = *(const v2f*)(arow + 4 * c);
  }

  float best[8];
  int   bidx[8];
#pragma unroll
  for (int i = 0; i < 8; ++i) { best[i] = -3.4e38f; bidx[i] = 0; }

  for (int g = 0; g < NCHUNK; ++g) {
    // wait for our async writes of chunk g, then make all waves' writes
    // visible (barrier also guarantees buffer (g+1)&1 is done being read)
    asm volatile("s_wait_asynccnt 0x0" ::: "memory");
    __syncthreads();
    if (g + 1 < NCHUNK)
      issue_chunk(W, g + 1, ldsW0 + ((g + 1) & 1) * (BUFF * 4u), threadIdx.x);

    const float* buf = &ldsW[(g & 1) * BUFF];
#pragma unroll
    for (int t = 0; t < CHUNK / 16; ++t) {       // 4 column tiles per chunk
      const int    nl = t * 16 + M;              // code row within chunk
      const float* bp = buf + nl * BROW + off;
      v8f acc = {};
#pragma unroll
      for (int c = 0; c < 64; c += 8) {          // 8 ds-loads in flight / 8 WMMAs
        v2f b0 = *(const v2f*)(bp + 4 * (c + 0));
        v2f b1 = *(const v2f*)(bp + 4 * (c + 1));
        v2f b2 = *(const v2f*)(bp + 4 * (c + 2));
        v2f b3 = *(const v2f*)(bp + 4 * (c + 3));
        v2f b4 = *(const v2f*)(bp + 4 * (c + 4));
        v2f b5 = *(const v2f*)(bp + 4 * (c + 5));
        v2f b6 = *(const v2f*)(bp + 4 * (c + 6));
        v2f b7 = *(const v2f*)(bp + 4 * (c + 7));
        acc = __builtin_amdgcn_wmma_f32_16x16x4_f32(false, a[c + 0], false, b0,
                                                    (short)0, acc, false, false);
        acc = __builtin_amdgcn_wmma_f32_16x16x4_f32(false, a[c + 1], false, b1,
                                                    (short)0, acc, false, false);
        acc = __builtin_amdgcn_wmma_f32_16x16x4_f32(false, a[c + 2], false, b2,
                                                    (short)0, acc, false, false);
        acc = __builtin_amdgcn_wmma_f32_16x16x4_f32(false, a[c + 3], false, b3,
                                                    (short)0, acc, false, false);
        acc = __builtin_amdgcn_wmma_f32_16x16x4_f32(false, a[c + 4], false, b4,
                                                    (short)0, acc, false, false);
        acc = __builtin_amdgcn_wmma_f32_16x16x4_f32(false, a[c + 5], false, b5,
                                                    (short)0, acc, false, false);
        acc = __builtin_amdgcn_wmma_f32_16x16x4_f32(false, a[c + 6], false, b6,
                                                    (short)0, acc, false, false);
        acc = __builtin_amdgcn_wmma_f32_16x16x4_f32(false, a[c + 7], false, b7,
                                                    (short)0, acc, false, false);
      }
      const int   n  = g * CHUNK + t * 16 + M;   // global code column
      const float sb = bias[n];                  // -0.5*||w_n||^2
#pragma unroll
      for (int i = 0; i < 8; ++i) {
        float s = acc[i] + sb;
        if (s > best[i]) { best[i] = s; bidx[i] = n; }
      }
    }
  }

  // --- argmax butterfly over the 16 lanes sharing a row (masks<=8 stay in half)
#pragma unroll
  for (int i = 0; i < 8; ++i) {
#pragma unroll
    for (int m = 8; m >= 1; m >>= 1) {
      float ob = __shfl_xor(best[i], m, 32);
      int   oi = __shfl_xor(bidx[i], m, 32);
      if (ob > best[i] || (ob == best[i] && oi < bidx[i])) {
        best[i] = ob; bidx[i] = oi;
      }
    }
  }
  // lanes 0-15 hold winner of row i; lanes 16-31 hold winner of row i+8
  if (lane == 0 || lane == 16) {
    const int half = (lane >> 4) << 3;
#pragma unroll
    for (int i = 0; i < 8; ++i) {
      kIdxS[wave][half + i] = bidx[i];
      out[IDX_OFF + rowBase + half + i] = (float)bidx[i];
    }
  }
  __syncthreads();

  // --- store z_q transposed [b, e, l] + accumulate (z_q - z_e)^2
  const int b  = rowBase / LPOS;
  const int l0 = rowBase % LPOS;                 // 16 consecutive l, same batch
  const int j  = lane & 15;
  const int eh = lane >> 4;
  const int myk = kIdxS[wave][j];
  const float* wrow = W + (size_t)myk * EMB;
  const float* xrow = X + (size_t)(rowBase + j) * EMB;
  float* obase = out + (size_t)b * EMB * LPOS + l0 + j;

  float local = 0.0f;
#pragma unroll 4
  for (int e = eh; e < EMB; e += 2) {
    float wv = wrow[e];
    float xv = xrow[e];
    float d  = wv - xv;
    local += d * d;
    obase[(size_t)e * LPOS] = wv;
  }
#pragma unroll
  for (int m = 16; m >= 1; m >>= 1) local += __shfl_xor(local, m, 32);
  if (lane == 0) redS[wave] = local;
  __syncthreads();
  if (threadIdx.x == 0) {
    float s = 0.0f;
#pragma unroll
    for (int w = 0; w < 8; ++w) s += redS[w];
    atomicAdd(diff_acc, s);
  }
}

// ---------------------------------------------------------------------------
// Finalize: diff = KLD_SCALE*(1+BETA)*mean = 12.5 * acc / (NROWS*EMB)
// ---------------------------------------------------------------------------
__global__ void vq_finalize(const float* __restrict__ ws,
                            float* __restrict__ out) {
  out[DIFF_OFF] = 12.5f * ws[0] / (float)((size_t)NROWS * EMB);
}

extern "C" void kernel_launch(void* const* d_in, const int* in_sizes, int n_in,
                              void* d_out, int out_size, void* d_ws, size_t ws_size,
                              hipStream_t stream) {
  const float* X  = (const float*)d_in[0];   // z_e  [16,2048,256] f32
  const float* W  = (const float*)d_in[1];   // embed_weight [1024,256] f32
  float* out = (float*)d_out;
  float* ws  = (float*)d_ws;                 // ws[0]=diff acc, ws[1..1024]=bias

  vq_prep<<<NCODE / 256, 256, 0, stream>>>(W, ws);
  vq_main<<<NROWS / 128, 256, 0, stream>>>(X, W, ws + 1, out, ws);
  vq_finalize<<<1, 1, 0, stream>>>(ws, out);
}